// Attacker_81570018886156
// MI455X (gfx1250) — compile-verified
//
#include <hip/hip_runtime.h>
#include <hip/hip_bf16.h>

typedef __attribute__((ext_vector_type(2))) float v2f;
typedef __attribute__((ext_vector_type(8))) float v8f;

#define F_IN 128
#define H1   180
#define H2   120
#define N_CLS 16
#define NEG_SLOPE 0.01f

// ---------------- degree / norm ----------------
__global__ __launch_bounds__(256) void k_deg_init(float* __restrict__ deg, int n) {
  int i = blockIdx.x * 256 + threadIdx.x;
  if (i < n) deg[i] = 1.0f;                      // self-loop
}

__global__ __launch_bounds__(256) void k_deg_count(const int* __restrict__ dst,
                                                   float* __restrict__ deg, int ne) {
  int e = blockIdx.x * 256 + threadIdx.x;
  if (e < ne) atomicAdd(&deg[dst[e]], 1.0f);
}

__global__ __launch_bounds__(256) void k_rsqrt(float* __restrict__ deg, int n) {
  int i = blockIdx.x * 256 + threadIdx.x;
  if (i < n) deg[i] = rsqrtf(deg[i]);
}

// ---------------- WMMA fp32 GEMM: C[M x N] = A[M x K] @ B[K x N] ----------------
// One wave per 16-row strip; NT = ceil(N/16) tiles.
// B is staged zero-padded + pre-swizzled in LDS so each lane's
// (B[k+koff][n], B[k+koff+1][n]) pair is a single aligned ds_load_b64,
// and the hot loop has no bounds checks / EXEC manipulation.
template <int K, int N, int NT>
__global__ __launch_bounds__(256) void k_wmma_gemm(const float* __restrict__ A,
                                                   const float* __restrict__ B,
                                                   float* __restrict__ C, int M) {
  constexpr int QU = K / 4;            // k-quads
  constexpr int NP = NT * 16;          // padded N
  constexpr int RS = NP * 2 + 2;       // floats per (q,h) LDS row (8B-aligned, skewed)
  __shared__ float bs[QU * 2 * RS];

  // cooperative staged load of B (zero-padded), swizzled into (q,h,n) pairs
  for (int idx = threadIdx.x; idx < QU * 2 * NP; idx += 256) {
    int n  = idx % NP;
    int qh = idx / NP;                         // q*2 + h
    int kr = (qh >> 1) * 4 + (qh & 1) * 2;     // source row = 4q + 2h
    float e0 = 0.0f, e1 = 0.0f;
    if (n < N) {
      e0 = B[(size_t)kr * N + n];
      e1 = B[(size_t)(kr + 1) * N + n];
    }
    bs[qh * RS + 2 * n]     = e0;
    bs[qh * RS + 2 * n + 1] = e1;
  }
  __syncthreads();

  int wave = (blockIdx.x * 256 + threadIdx.x) >> 5;
  int lane = threadIdx.x & 31;
  int m0 = wave * 16;
  if (m0 < M) {
    const int mrow = lane & 15;          // A: M index within tile
    const int koff = (lane >> 4) << 1;   // A: +0 or +2 within k-quad
    const int hsel = lane >> 4;          // B: which (q,h) LDS row
    const int ncol = lane & 15;          // B/C: N index within tile

    v8f acc[NT] = {};
    int arowi = m0 + mrow; if (arowi >= M) arowi = M - 1;   // safe (M%16==0 normally)
    const float* __restrict__ arow = A + (size_t)arowi * K;

    for (int q = 0; q < QU; ++q) {
      v2f a;
      a.x = arow[q * 4 + koff];
      a.y = arow[q * 4 + koff + 1];
      const float* __restrict__ brow = &bs[(q * 2 + hsel) * RS];
#pragma unroll
      for (int t = 0; t < NT; ++t) {
        v2f b = *(const v2f*)&brow[2 * (t * 16 + ncol)];
        acc[t] = __builtin_amdgcn_wmma_f32_16x16x4_f32(false, a, false, b,
                                                       (short)0, acc[t], false, false);
      }
    }

    // C/D layout: vgpr v, lanes 0-15 -> M = v; lanes 16-31 -> M = v+8; N = lane%16
    int mbase = m0 + (lane >> 4) * 8;
#pragma unroll
    for (int t = 0; t < NT; ++t) {
      int n = t * 16 + ncol;
      if (n < N) {
#pragma unroll
        for (int v = 0; v < 8; ++v) {
          int m = mbase + v;
          if (m < M) C[(size_t)m * N + n] = acc[t][v];
        }
      }
    }
  }
}

// ---------------- aggregation ----------------
template <int C>
__global__ __launch_bounds__(256) void k_self_init(const float* __restrict__ h,
                                                   const float* __restrict__ dinv,
                                                   float* __restrict__ out, int total) {
  int i = blockIdx.x * 256 + threadIdx.x;
  if (i < total) {
    int node = i / C;
    float di = dinv[node];
    out[i] = h[i] * di * di;                     // self-loop message
  }
}

template <int C>
__global__ __launch_bounds__(256) void k_edge_scatter(const float* __restrict__ h,
                                                      const int* __restrict__ src,
                                                      const int* __restrict__ dst,
                                                      const float* __restrict__ dinv,
                                                      float* __restrict__ out, int ne) {
  int wave = (blockIdx.x * 256 + threadIdx.x) >> 5;   // one wave32 per edge
  int lane = threadIdx.x & 31;
  if (wave >= ne) return;
  int s = src[wave], d = dst[wave];
  float norm = dinv[s] * dinv[d];
  const float* __restrict__ hp = h + (size_t)s * C;   // coalesced gather
  float* __restrict__ op = out + (size_t)d * C;       // coalesced atomics
  for (int c = lane; c < C; c += 32)
    atomicAdd(&op[c], hp[c] * norm);
}

template <int C>
__global__ __launch_bounds__(256) void k_bias_act(float* __restrict__ x,
                                                  const float* __restrict__ b, int total) {
  int i = blockIdx.x * 256 + threadIdx.x;
  if (i < total) {
    float v = x[i] + b[i % C];
    x[i] = (v > 0.0f) ? v : v * NEG_SLOPE;       // leaky relu
  }
}

// ---------------- bias + log_softmax over 16 classes ----------------
__global__ __launch_bounds__(256) void k_logsoftmax16(float* __restrict__ out,
                                                      const float* __restrict__ bl, int n) {
  int r = blockIdx.x * 256 + threadIdx.x;
  if (r >= n) return;
  float* __restrict__ p = out + (size_t)r * N_CLS;
  float v[N_CLS];
  float mx = -3.4e38f;
#pragma unroll
  for (int c = 0; c < N_CLS; ++c) { v[c] = p[c] + bl[c]; mx = fmaxf(mx, v[c]); }
  float s = 0.0f;
#pragma unroll
  for (int c = 0; c < N_CLS; ++c) s += expf(v[c] - mx);
  float ls = mx + logf(s);
#pragma unroll
  for (int c = 0; c < N_CLS; ++c) p[c] = v[c] - ls;
}

extern "C" void kernel_launch(void* const* d_in, const int* in_sizes, int n_in,
                              void* d_out, int out_size, void* d_ws, size_t ws_size,
                              hipStream_t stream) {
  (void)n_in; (void)out_size; (void)ws_size;
  const float* x  = (const float*)d_in[0];
  const float* W1 = (const float*)d_in[1];
  const float* b1 = (const float*)d_in[2];
  const float* W2 = (const float*)d_in[3];
  const float* b2 = (const float*)d_in[4];
  const float* Wl = (const float*)d_in[5];
  const float* bl = (const float*)d_in[6];
  const int*   ei = (const int*)d_in[7];

  const int nE = in_sizes[7] / 2;
  const int nN = in_sizes[0] / F_IN;
  const int* src = ei;
  const int* dst = ei + nE;

  // workspace layout: [dinv | bufA (nN*H1) | bufB (nN*H1)]
  char* ws = (char*)d_ws;
  float* dinv = (float*)ws;
  float* bufA = (float*)(ws + (size_t)512 * 1024);
  float* bufB = bufA + (size_t)nN * H1;
  float* outp = (float*)d_out;

  const int nWaves = (nN + 15) / 16;
  const int gemmBlocks = (nWaves * 32 + 255) / 256;

  // normalization coefficients
  k_deg_init <<<(nN + 255) / 256, 256, 0, stream>>>(dinv, nN);
  k_deg_count<<<(nE + 255) / 256, 256, 0, stream>>>(dst, dinv, nE);
  k_rsqrt    <<<(nN + 255) / 256, 256, 0, stream>>>(dinv, nN);

  // ---- layer 1: h1 = x @ W1 ; agg ; +b1 ; leaky ----
  k_wmma_gemm<F_IN, H1, 12><<<gemmBlocks, 256, 0, stream>>>(x, W1, bufA, nN);
  {
    int tot = nN * H1;
    k_self_init<H1>   <<<(tot + 255) / 256, 256, 0, stream>>>(bufA, dinv, bufB, tot);
    k_edge_scatter<H1><<<((size_t)nE * 32 + 255) / 256, 256, 0, stream>>>(bufA, src, dst, dinv, bufB, nE);
    k_bias_act<H1>    <<<(tot + 255) / 256, 256, 0, stream>>>(bufB, b1, tot);
  }

  // ---- layer 2: h2 = a1 @ W2 ; agg ; +b2 ; leaky ----
  k_wmma_gemm<H1, H2, 8><<<gemmBlocks, 256, 0, stream>>>(bufB, W2, bufA, nN);
  {
    int tot = nN * H2;
    k_self_init<H2>   <<<(tot + 255) / 256, 256, 0, stream>>>(bufA, dinv, bufB, tot);
    k_edge_scatter<H2><<<((size_t)nE * 32 + 255) / 256, 256, 0, stream>>>(bufA, src, dst, dinv, bufB, nE);
    k_bias_act<H2>    <<<(tot + 255) / 256, 256, 0, stream>>>(bufB, b2, tot);
  }

  // ---- head: logits = a2 @ Wl ; +bl ; log_softmax ----
  k_wmma_gemm<H2, N_CLS, 1><<<gemmBlocks, 256, 0, stream>>>(bufB, Wl, outp, nN);
  k_logsoftmax16<<<(nN + 255) / 256, 256, 0, stream>>>(outp, bl, nN);
}